// SAGE_27212912787987
// MI455X (gfx1250) — compile-verified
//
#include <hip/hip_runtime.h>

typedef float v2f __attribute__((ext_vector_type(2)));
typedef float v8f __attribute__((ext_vector_type(8)));

#define N_NODES 50000
#define N_EDGES 600000
#define F_IN 128
#define HID 128
#define N_CLS 40

// ---------------------------------------------------------------------------
// Zero a float region (grid-stride).
// ---------------------------------------------------------------------------
__global__ void sage_zero_f32(float* __restrict__ p, long n) {
  long i = (long)blockIdx.x * blockDim.x + threadIdx.x;
  long stride = (long)gridDim.x * blockDim.x;
  for (; i < n; i += stride) p[i] = 0.0f;
}

// ---------------------------------------------------------------------------
// Per-destination in-degree via float atomics (one thread per edge).
// ---------------------------------------------------------------------------
__global__ void sage_count_edges(const int* __restrict__ dst,
                                 float* __restrict__ cnt) {
  int e = blockIdx.x * blockDim.x + threadIdx.x;
  if (e < N_EDGES) atomicAdd(&cnt[dst[e]], 1.0f);
}

// ---------------------------------------------------------------------------
// Scatter-add of 128-float feature rows: one wave32 per edge, 4 floats/lane.
// feat is [N x 128] row-major; sum is [N x 128] accumulated with f32 atomics.
// ---------------------------------------------------------------------------
__global__ void sage_scatter_feat(const float* __restrict__ feat,
                                  const int* __restrict__ src,
                                  const int* __restrict__ dst,
                                  float* __restrict__ sum) {
  int wave = (int)((blockIdx.x * (long)blockDim.x + threadIdx.x) >> 5);
  int lane = threadIdx.x & 31;
  if (wave >= N_EDGES) return;
  int s = src[wave];
  int d = dst[wave];
  const float4 v = *(const float4*)(feat + (long)s * F_IN + lane * 4);
  float* o = sum + (long)d * F_IN + lane * 4;
  atomicAdd(o + 0, v.x);
  atomicAdd(o + 1, v.y);
  atomicAdd(o + 2, v.z);
  atomicAdd(o + 3, v.w);
}

// ---------------------------------------------------------------------------
// Fused dual-GEMM SAGE dense stage using V_WMMA_F32_16X16X4_F32 (exact fp32):
//   out = act( (aggsum/max(cnt,1)) @ wl + xin @ wr + bias )
//
// Block = 256 threads = 8 waves; all waves share one N-tile (nt) and stage
// that tile's weight columns for BOTH matrices into LDS once (16 KB of the
// 320 KB/WGP pool). Inner loop then feeds B fragments from ds_load with no
// bank conflicts (lanes 0-15: banks kk*16..+15; lanes 16-31: (kk+2)*16..+15,
// disjoint mod 64) and A fragments from global. Tail M-tiles are handled by
// clamping mt (not early-return) so every wave reaches the barrier and all
// WMMAs execute with EXEC all-1s; only the store is masked.
//
// f32 WMMA VGPR layouts (CDNA5 ISA):
//   A 16x4:  lane holds row = lane%16, K pair = (lane/16)*2 .. +1   -> v2f
//   B 4x16:  lane holds col = lane%16, K pair = (lane/16)*2 .. +1   -> v2f
//   C 16x16: VGPR r: row = (lane/16)*8 + r, col = lane%16           -> v8f
// ---------------------------------------------------------------------------
template <int K, int OUT, bool RELU>
__global__ void sage_dense_wmma(const float* __restrict__ xin,    // [N x K]
                                const float* __restrict__ aggsum, // [N x K]
                                const float* __restrict__ cnt,    // [N]
                                const float* __restrict__ wl,     // [K x OUT]
                                const float* __restrict__ wr,     // [K x OUT]
                                const float* __restrict__ bias,   // [OUT]
                                float* __restrict__ out) {        // [N x OUT]
  constexpr int NT = (OUT + 15) / 16;       // N tiles (padded)
  constexpr int MT = N_NODES / 16;          // 3125 exact

  __shared__ float lwl[K][16];              // wl[:, nt*16 .. +16]
  __shared__ float lwr[K][16];              // wr[:, nt*16 .. +16]

  const int nt = blockIdx.x % NT;
  const int mtBase = (blockIdx.x / NT) * 8;
  const int waveInBlk = threadIdx.x >> 5;
  const int lane = threadIdx.x & 31;

  // Cooperative weight staging: zero-pad columns beyond OUT so the inner
  // loop is fully divergence-free even for the OUT=40 tail tile.
  for (int i = threadIdx.x; i < K * 16; i += 256) {
    const int k = i >> 4;
    const int c = i & 15;
    const int col = nt * 16 + c;
    const bool ok = (OUT % 16 == 0) || (col < OUT);
    lwl[k][c] = ok ? wl[(long)k * OUT + col] : 0.0f;
    lwr[k][c] = ok ? wr[(long)k * OUT + col] : 0.0f;
  }
  __syncthreads();

  int mt = mtBase + waveInBlk;
  const bool mOK = mt < MT;
  if (!mOK) mt = MT - 1;                    // clamp: keep wave alive for WMMA

  const int halfsel = lane >> 4;            // 0: K+0..1, 1: K+2..3
  const int l16 = lane & 15;
  const int arow = mt * 16 + l16;           // A row held by this lane
  const int bcol = nt * 16 + l16;           // B col held by this lane

  const float inv = 1.0f / fmaxf(cnt[arow], 1.0f);
  const float* __restrict__ xrow = xin + (long)arow * K;
  const float* __restrict__ srow = aggsum + (long)arow * K;

  v8f c = {0.f, 0.f, 0.f, 0.f, 0.f, 0.f, 0.f, 0.f};

  for (int k0 = 0; k0 < K; k0 += 4) {
    const int kk = k0 + halfsel * 2;        // even -> 8B aligned
    v2f am = *(const v2f*)(srow + kk);
    am.x *= inv;
    am.y *= inv;
    const v2f ax = *(const v2f*)(xrow + kk);

    v2f bl, br;
    bl.x = lwl[kk][l16];
    bl.y = lwl[kk + 1][l16];
    br.x = lwr[kk][l16];
    br.y = lwr[kk + 1][l16];

    // D = A*B + C in exact fp32 on the matrix pipe.
    c = __builtin_amdgcn_wmma_f32_16x16x4_f32(false, am, false, bl,
                                              (short)0, c, false, false);
    c = __builtin_amdgcn_wmma_f32_16x16x4_f32(false, ax, false, br,
                                              (short)0, c, false, false);
  }

  const bool colOK = (OUT % 16 == 0) || (bcol < OUT);
  if (!mOK || !colOK) return;               // divergence only after all WMMAs
  const float bv = bias[bcol];
#pragma unroll
  for (int r = 0; r < 8; ++r) {
    const int row = mt * 16 + halfsel * 8 + r;
    float v = c[r] + bv;
    if (RELU) v = fmaxf(v, 0.0f);
    out[(long)row * OUT + bcol] = v;
  }
}

// ---------------------------------------------------------------------------
// Host-side orchestration (stream-only; graph-capture safe).
// ---------------------------------------------------------------------------
extern "C" void kernel_launch(void* const* d_in, const int* in_sizes, int n_in,
                              void* d_out, int out_size, void* d_ws, size_t ws_size,
                              hipStream_t stream) {
  const float* x    = (const float*)d_in[0];                 // [50000 x 128]
  const int*   ei   = (const int*)d_in[1];                   // [2 x 600000]
  const float* w1l  = (const float*)d_in[2];                 // [128 x 128]
  const float* w1r  = (const float*)d_in[3];                 // [128 x 128]
  const float* b1   = (const float*)d_in[4];                 // [128]
  const float* w2l  = (const float*)d_in[5];                 // [128 x 40]
  const float* w2r  = (const float*)d_in[6];                 // [128 x 40]
  const float* b2   = (const float*)d_in[7];                 // [40]
  float* out = (float*)d_out;                                // [50000 x 40]

  const int* src = ei;
  const int* dst = ei + N_EDGES;

  // Workspace carve-up (sum | cnt | h).
  const long sumElems = (long)N_NODES * F_IN;                // 6.4M floats
  float* sum = (float*)d_ws;
  float* cnt = sum + sumElems;                               // 50000 floats
  float* h   = cnt + ((N_NODES + 63) & ~63);                 // [50000 x 128]

  const dim3 blk(256);

  // 1) zero sum + cnt (contiguous region)
  {
    long n = sumElems + ((N_NODES + 63) & ~63);
    sage_zero_f32<<<2048, blk, 0, stream>>>(sum, n);
  }
  // 2) in-degree counts
  sage_count_edges<<<(N_EDGES + 255) / 256, blk, 0, stream>>>(dst, cnt);
  // 3) scatter x into sum (one wave per edge)
  sage_scatter_feat<<<(N_EDGES * 32 + 255) / 256, blk, 0, stream>>>(x, src, dst, sum);
  // 4) layer 1: h = relu(mean @ w1l + x @ w1r + b1)
  {
    const int NT = HID / 16;                                 // 8
    const int blocks = ((N_NODES / 16 + 7) / 8) * NT;        // 391 * 8
    sage_dense_wmma<F_IN, HID, true>
        <<<blocks, blk, 0, stream>>>(x, sum, cnt, w1l, w1r, b1, h);
  }
  // 5) re-zero sum only (cnt is reused)
  sage_zero_f32<<<2048, blk, 0, stream>>>(sum, sumElems);
  // 6) scatter h into sum
  sage_scatter_feat<<<(N_EDGES * 32 + 255) / 256, blk, 0, stream>>>(h, src, dst, sum);
  // 7) layer 2: out = mean @ w2l + h @ w2r + b2
  {
    const int NT = (N_CLS + 15) / 16;                        // 3
    const int blocks = ((N_NODES / 16 + 7) / 8) * NT;        // 391 * 3
    sage_dense_wmma<HID, N_CLS, false>
        <<<blocks, blk, 0, stream>>>(h, sum, cnt, w2l, w2r, b2, out);
  }
}